// GENConvTwoDirection_63479616635260
// MI455X (gfx1250) — compile-verified
//
#include <hip/hip_runtime.h>

// ---------------------------------------------------------------------------
// GENConv two-direction (softmax aggregation, learn_t, 2-layer MLP w/ LN)
// MI455X / gfx1250: f32 WMMA (V_WMMA_F32_16X16X4_F32) for all GEMMs.
// ---------------------------------------------------------------------------

typedef __attribute__((ext_vector_type(2))) float v2f;
typedef __attribute__((ext_vector_type(8))) float v8f;

#define GEN_EPS   1e-7f
#define LN_EPS    1e-5f
#define CDIM      64
#define LEFT_DIM  48
#define RIGHT_DIM 56

// ---- order-preserving float <-> uint encoding (for atomicMax on floats) ----
__device__ __forceinline__ unsigned flt_enc(float f) {
    unsigned u = __float_as_uint(f);
    return (u & 0x80000000u) ? ~u : (u | 0x80000000u);
}
__device__ __forceinline__ float flt_dec(unsigned u) {
    return (u & 0x80000000u) ? __uint_as_float(u & 0x7fffffffu)
                             : __uint_as_float(~u);
}

// ---- one 16x16x4 f32 WMMA step -------------------------------------------
// A: 16 rows x lda (row-major, LDS), rows = lane&15.
//    ISA A layout: VGPR0 = K {k, k+2} (lane half selects), VGPR1 = K {k+1, k+3}.
// B: K x ldb row-major in global (L2-resident weights); VGPR0 = rows {k, k+2},
//    VGPR1 = rows {k+1, k+3}, column = n0 + (lane&15).
__device__ __forceinline__ v8f wmma4(v8f c, const float* __restrict__ A, int lda,
                                     const float* __restrict__ B, int ldb,
                                     int k, int n0, int m, int kh) {
    const int kk = k + 2 * kh;
    v2f a, b;
    a.x = A[m * lda + kk];
    a.y = A[m * lda + kk + 1];
    b.x = B[kk * ldb + n0 + m];
    b.y = B[(kk + 1) * ldb + n0 + m];
    return __builtin_amdgcn_wmma_f32_16x16x4_f32(false, a, false, b,
                                                 (short)0, c, false, false);
}

// ---------------------------------------------------------------------------
// h_src[n,c] = sum_k X[n,k] * W[k,c]   (N multiple of 16, K multiple of 4)
// one wave (32 threads) per 16-row tile
// ---------------------------------------------------------------------------
template <int K>
__global__ void gemm_hsrc_kernel(const float* __restrict__ X,
                                 const float* __restrict__ W,
                                 float* __restrict__ H) {
    __shared__ float sX[16 * K];
    const int tile = blockIdx.x;
    const int lane = threadIdx.x;
    const float* Xt = X + (long long)tile * 16 * K;
    for (int i = lane; i < 16 * K; i += 32) sX[i] = Xt[i];
    __syncthreads();

    const int m = lane & 15;
    const int kh = lane >> 4;
    float* Ht = H + (long long)tile * 16 * CDIM;
    for (int ct = 0; ct < 4; ++ct) {
        v8f acc = {};
        for (int k = 0; k < K; k += 4)
            acc = wmma4(acc, sX, K, W, CDIM, k, ct * 16, m, kh);
#pragma unroll
        for (int j = 0; j < 8; ++j)
            Ht[(j + 8 * kh) * CDIM + ct * 16 + m] = acc[j];
    }
}

// ---------------------------------------------------------------------------
// Edge pass 1: segment max of logits (encoded uint atomicMax)
// thread -> (edge, channel); 64 channels per edge
// ---------------------------------------------------------------------------
__global__ void edge_max_kernel(const float* __restrict__ hsrc,
                                const float* __restrict__ eval,
                                const int* __restrict__ src,
                                const int* __restrict__ dst,
                                const float* __restrict__ we,
                                const float* __restrict__ t_ptr,
                                unsigned* __restrict__ mx_enc, long long E) {
    const long long gid = (long long)blockIdx.x * blockDim.x + threadIdx.x;
    const long long e = gid >> 6;
    const int c = (int)(gid & 63);
    if (e >= E) return;
    const float t = t_ptr[0];
    float msg = hsrc[(long long)src[e] * CDIM + c] + eval[e] * we[c];
    msg = fmaxf(msg, 0.0f) + GEN_EPS;
    const float logit = msg * t;
    atomicMax(&mx_enc[(long long)dst[e] * CDIM + c], flt_enc(logit));
}

// ---------------------------------------------------------------------------
// Edge pass 2: denom += ex, num += msg*ex   (agg = num / (denom + 1e-16))
// ---------------------------------------------------------------------------
__global__ void edge_sum_kernel(const float* __restrict__ hsrc,
                                const float* __restrict__ eval,
                                const int* __restrict__ src,
                                const int* __restrict__ dst,
                                const float* __restrict__ we,
                                const float* __restrict__ t_ptr,
                                const unsigned* __restrict__ mx_enc,
                                float* __restrict__ den,
                                float* __restrict__ num, long long E) {
    const long long gid = (long long)blockIdx.x * blockDim.x + threadIdx.x;
    const long long e = gid >> 6;
    const int c = (int)(gid & 63);
    if (e >= E) return;
    const float t = t_ptr[0];
    float msg = hsrc[(long long)src[e] * CDIM + c] + eval[e] * we[c];
    msg = fmaxf(msg, 0.0f) + GEN_EPS;
    const float logit = msg * t;
    const long long off = (long long)dst[e] * CDIM + c;
    const float mx = flt_dec(mx_enc[off]);
    const float ex = __expf(logit - mx);
    atomicAdd(&den[off], ex);
    atomicAdd(&num[off], msg * ex);
}

// ---------------------------------------------------------------------------
// Node finalize: out = agg + Xd@Wd ; y = relu(LN(out@W1)) @ W2
// one wave per 16-node tile, all GEMMs via f32 WMMA
// ---------------------------------------------------------------------------
template <int KD>
__global__ void node_update_kernel(const float* __restrict__ Xd,
                                   const float* __restrict__ num,
                                   const float* __restrict__ den,
                                   const float* __restrict__ Wd,
                                   const float* __restrict__ W1,
                                   const float* __restrict__ ln_g,
                                   const float* __restrict__ ln_b,
                                   const float* __restrict__ W2,
                                   float* __restrict__ Y) {
    __shared__ float sX[16 * KD];     // x_dst tile
    __shared__ float sA[16 * CDIM];   // out = agg + xd@Wd
    __shared__ float sH[16 * 128];    // hidden (pre/post LN)

    const int tile = blockIdx.x;
    const int lane = threadIdx.x;
    const float* Xt = Xd + (long long)tile * 16 * KD;
    for (int i = lane; i < 16 * KD; i += 32) sX[i] = Xt[i];
    __syncthreads();

    const int m = lane & 15;
    const int kh = lane >> 4;
    const long long base = (long long)tile * 16 * CDIM;

    // out = agg + Xd @ Wd
    for (int ct = 0; ct < 4; ++ct) {
        v8f acc = {};
        for (int k = 0; k < KD; k += 4)
            acc = wmma4(acc, sX, KD, Wd, CDIM, k, ct * 16, m, kh);
#pragma unroll
        for (int j = 0; j < 8; ++j) {
            const int row = j + 8 * kh;
            const int col = ct * 16 + m;
            const long long idx = base + row * CDIM + col;
            const float agg = num[idx] / (den[idx] + 1e-16f);
            sA[row * CDIM + col] = acc[j] + agg;
        }
    }
    __syncthreads();

    // h1 = out @ W1  (64 -> 128)
    for (int nt = 0; nt < 8; ++nt) {
        v8f acc = {};
        for (int k = 0; k < CDIM; k += 4)
            acc = wmma4(acc, sA, CDIM, W1, 128, k, nt * 16, m, kh);
#pragma unroll
        for (int j = 0; j < 8; ++j)
            sH[(j + 8 * kh) * 128 + nt * 16 + m] = acc[j];
    }
    __syncthreads();

    // LayerNorm (population variance) + ReLU, one lane per row
    if (lane < 16) {
        float s = 0.0f, s2 = 0.0f;
        for (int c = 0; c < 128; ++c) {
            const float v = sH[lane * 128 + c];
            s += v;
            s2 += v * v;
        }
        const float mean = s * (1.0f / 128.0f);
        const float var = s2 * (1.0f / 128.0f) - mean * mean;
        const float rstd = rsqrtf(var + LN_EPS);
        for (int c = 0; c < 128; ++c) {
            const float v =
                (sH[lane * 128 + c] - mean) * rstd * ln_g[c] + ln_b[c];
            sH[lane * 128 + c] = fmaxf(v, 0.0f);
        }
    }
    __syncthreads();

    // y = h @ W2  (128 -> 64)
    for (int ot = 0; ot < 4; ++ot) {
        v8f acc = {};
        for (int k = 0; k < 128; k += 4)
            acc = wmma4(acc, sH, 128, W2, CDIM, k, ot * 16, m, kh);
#pragma unroll
        for (int j = 0; j < 8; ++j)
            Y[base + (j + 8 * kh) * CDIM + ot * 16 + m] = acc[j];
    }
}

// ---------------------------------------------------------------------------
// Host side
// ---------------------------------------------------------------------------
struct GenParams {
    const float *lin_src, *lin_edge, *lin_dst, *t, *w1, *ln_g, *ln_b, *w2;
};

extern "C" void kernel_launch(void* const* d_in, const int* in_sizes, int n_in,
                              void* d_out, int out_size, void* d_ws,
                              size_t ws_size, hipStream_t stream) {
    const float* left  = (const float*)d_in[0];
    const float* right = (const float*)d_in[1];
    const float* eval  = (const float*)d_in[2];
    const int* esrc = (const int*)d_in[n_in - 2];
    const int* edst = (const int*)d_in[n_in - 1];

    const int NL = in_sizes[0] / LEFT_DIM;   // 100000
    const int NR = in_sizes[1] / RIGHT_DIM;  // 100000
    const long long E = in_sizes[2];         // 1000000

    // Param-dict flattening order is either insertion order
    // (lin_src, lin_edge, lin_dst, t, mlp_w1, ln_g, ln_b, mlp_w2) or
    // key-sorted (lin_dst, lin_edge, lin_src, ln_b, ln_g, mlp_w1, mlp_w2, t).
    // Leaf sizes distinguish them: the scalar t sits at +3 (insertion) vs +7.
    auto load_params = [&](int b) -> GenParams {
        GenParams p;
        if (in_sizes[b + 3] == 1) {  // insertion order
            p.lin_src  = (const float*)d_in[b + 0];
            p.lin_edge = (const float*)d_in[b + 1];
            p.lin_dst  = (const float*)d_in[b + 2];
            p.t        = (const float*)d_in[b + 3];
            p.w1       = (const float*)d_in[b + 4];
            p.ln_g     = (const float*)d_in[b + 5];
            p.ln_b     = (const float*)d_in[b + 6];
            p.w2       = (const float*)d_in[b + 7];
        } else {  // alphabetical
            p.lin_dst  = (const float*)d_in[b + 0];
            p.lin_edge = (const float*)d_in[b + 1];
            p.lin_src  = (const float*)d_in[b + 2];
            p.ln_b     = (const float*)d_in[b + 3];
            p.ln_g     = (const float*)d_in[b + 4];
            p.w1       = (const float*)d_in[b + 5];
            p.w2       = (const float*)d_in[b + 6];
            p.t        = (const float*)d_in[b + 7];
        }
        return p;
    };
    const GenParams pl2r = load_params(3);
    const GenParams pr2l = load_params(11);

    float* out_left  = (float*)d_out;                      // left_updated
    float* out_right = (float*)d_out + (size_t)NL * CDIM;  // right_updated

    // Workspace: 4 buffers of NMAX*64 (h_src, mx, den, num), reused per dir.
    const size_t nmax = (size_t)((NL > NR) ? NL : NR);
    const size_t per = nmax * CDIM;
    float*    hsrc = (float*)d_ws;
    unsigned* mx   = (unsigned*)((char*)d_ws + per * 4);
    float*    den  = (float*)((char*)d_ws + per * 8);
    float*    num  = (float*)((char*)d_ws + per * 12);

    const long long etot = E * CDIM;
    const int eblocks = (int)((etot + 255) / 256);

    auto run_dir = [&](const float* xs, int ns, int ks, const float* xd, int nd,
                       int kd, const int* sidx, const int* didx,
                       const GenParams& p, float* y) {
        hipMemsetAsync(mx, 0, (size_t)nd * CDIM * 4, stream);
        hipMemsetAsync(den, 0, (size_t)nd * CDIM * 4, stream);
        hipMemsetAsync(num, 0, (size_t)nd * CDIM * 4, stream);
        if (ks == LEFT_DIM)
            gemm_hsrc_kernel<LEFT_DIM>
                <<<ns / 16, 32, 0, stream>>>(xs, p.lin_src, hsrc);
        else
            gemm_hsrc_kernel<RIGHT_DIM>
                <<<ns / 16, 32, 0, stream>>>(xs, p.lin_src, hsrc);
        edge_max_kernel<<<eblocks, 256, 0, stream>>>(hsrc, eval, sidx, didx,
                                                     p.lin_edge, p.t, mx, E);
        edge_sum_kernel<<<eblocks, 256, 0, stream>>>(
            hsrc, eval, sidx, didx, p.lin_edge, p.t, mx, den, num, E);
        if (kd == RIGHT_DIM)
            node_update_kernel<RIGHT_DIM><<<nd / 16, 32, 0, stream>>>(
                xd, num, den, p.lin_dst, p.w1, p.ln_g, p.ln_b, p.w2, y);
        else
            node_update_kernel<LEFT_DIM><<<nd / 16, 32, 0, stream>>>(
                xd, num, den, p.lin_dst, p.w1, p.ln_g, p.ln_b, p.w2, y);
    };

    // left -> right (output: right_updated), then right -> left
    run_dir(left, NL, LEFT_DIM, right, NR, RIGHT_DIM, esrc, edst, pl2r,
            out_right);
    run_dir(right, NR, RIGHT_DIM, left, NL, LEFT_DIM, edst, esrc, pr2l,
            out_left);
}